// DMXQuantizer_13924283973786
// MI455X (gfx1250) — compile-verified
//
#include <hip/hip_runtime.h>
#include <hip/hip_bf16.h>

// BFP8 block-floating-point cast, block = 128 elems along last dim.
// Memory-bound (512 MB @ 23.3 TB/s ~ 22 us). One wave32 per 128-elem block:
// lane -> float4, ds_swizzle butterfly for block amax, v_ldexp/v_rndne math,
// async global->LDS double-buffered pipeline for the streaming loads.

typedef __attribute__((ext_vector_type(4))) float v4f;

// The gfx1250 async-LDS builtin wants pointers to 4xi32 vectors with explicit
// address spaces: (v4i __device__* src, v4i __shared__* dst, imm off, imm cpol)
typedef __attribute__((__vector_size__(16))) int v4i_t;
typedef __attribute__((address_space(1))) v4i_t as1_v4i;
typedef __attribute__((address_space(3))) v4i_t as3_v4i;

#define THREADS      256
#define TILE_FLOATS  1024u   // 8 BFP blocks of 128 per workgroup tile (4 KB)
#define MAX_GRID     4096u

#if defined(__HIP_DEVICE_COMPILE__) && \
    __has_builtin(__builtin_amdgcn_global_load_async_to_lds_b128)
#define BFP_HAS_ASYNC 1
#else
#define BFP_HAS_ASYNC 0
#endif

#if defined(__HIP_DEVICE_COMPILE__) && __has_builtin(__builtin_amdgcn_s_wait_asynccnt)
#define WAIT_ASYNC(n) __builtin_amdgcn_s_wait_asynccnt(n)
#else
#define WAIT_ASYNC(n) asm volatile("s_wait_asynccnt " #n ::: "memory")
#endif

// wave32 butterfly max via DS_SWIZZLE group-of-32 (and=0x1F, or=0, xor=k<<10)
#define SWZ_MAX(m, pat) \
  fmaxf((m), __int_as_float(__builtin_amdgcn_ds_swizzle(__float_as_int(m), (pat))))

// exact floor(log2(m)) for m >= 0 via exponent bits; subnormals renormalized.
__device__ __forceinline__ int floor_log2_f32(float m) {
  const bool sub = m < 0x1p-126f;          // subnormal or zero amax
  const float mn = sub ? m * 0x1p64f : m;  // renormalize into normal range
  const int e = (int)((__float_as_uint(mn) >> 23) & 0xFFu) - 127;
  return sub ? e - 64 : e;
}

__device__ __forceinline__ v4f bfp8_quantize4(v4f v) {
  // per-lane partial max of |x| over 4 elems
  float m = fmaxf(fmaxf(fabsf(v.x), fabsf(v.y)), fmaxf(fabsf(v.z), fabsf(v.w)));
  // reduce across the 32 lanes of the wave (one 128-elem block)
  m = SWZ_MAX(m, 0x401F);  // xor 16
  m = SWZ_MAX(m, 0x201F);  // xor 8
  m = SWZ_MAX(m, 0x101F);  // xor 4
  m = SWZ_MAX(m, 0x081F);  // xor 2
  m = SWZ_MAX(m, 0x041F);  // xor 1

  const int e  = floor_log2_f32(m);   // shared exponent
  const int k1 = 6 - e;               // x / 2^(e-6) == ldexp(x, 6-e)
  const int k2 = e - 6;

  v4f r;
#pragma unroll
  for (int i = 0; i < 4; ++i) {
    float q = __builtin_rintf(__builtin_ldexpf(v[i], k1));  // RNE like jnp.round
    q = fminf(fmaxf(q, -127.0f), 127.0f);                   // symmetric clamp
    r[i] = __builtin_ldexpf(q, k2);
  }
  // zero block: m==0 -> all v[i]==0 -> q==0 -> r==0 (matches reference guard)
  return r;
}

__global__ __launch_bounds__(THREADS)
void bfp8_cast_kernel(const float* __restrict__ x, float* __restrict__ y,
                      unsigned n_blocks) {
  const unsigned tid     = threadIdx.x;          // 0..255, wave = tid>>5
  const unsigned n_tiles = (n_blocks + 7u) >> 3; // 8 blocks per tile
  const unsigned t0      = blockIdx.x;
  if (t0 >= n_tiles) return;

#if BFP_HAS_ASYNC
  __shared__ __align__(16) float lds[2][TILE_FLOATS];

  auto issue_tile = [&](unsigned tile, unsigned b) {
    if (tile * 8u + (tid >> 5) < n_blocks) {
      const float* gp = x + (size_t)tile * TILE_FLOATS + tid * 4u;
      __builtin_amdgcn_global_load_async_to_lds_b128(
          (as1_v4i*)const_cast<float*>(gp),
          (as3_v4i*)&lds[b][tid * 4u],
          /*offset=*/0, /*cpol=*/0);
    }
  };

  issue_tile(t0, 0u);                      // prologue: prime buffer 0
  unsigned buf = 0u;
  for (unsigned t = t0; t < n_tiles; t += gridDim.x) {
    const unsigned tn = t + gridDim.x;
    if (tn < n_tiles) {
      issue_tile(tn, buf ^ 1u);            // prefetch next tile into other buf
      WAIT_ASYNC(1);                       // oldest async (this tile) is done
    } else {
      WAIT_ASYNC(0);
    }
    asm volatile("" ::: "memory");         // keep LDS reads below the wait
    const unsigned blk = t * 8u + (tid >> 5);
    if (blk < n_blocks) {
      const v4f v = *(const v4f*)&lds[buf][tid * 4u];  // ds_load_b128
      const v4f r = bfp8_quantize4(v);
      *(v4f*)(y + (size_t)t * TILE_FLOATS + tid * 4u) = r;  // global_store_b128
    }
    buf ^= 1u;
  }
#else
  for (unsigned t = t0; t < n_tiles; t += gridDim.x) {
    const unsigned blk = t * 8u + (tid >> 5);
    if (blk < n_blocks) {
      const float* gp = x + (size_t)t * TILE_FLOATS + tid * 4u;
      const v4f v = *(const v4f*)gp;       // global_load_b128
      const v4f r = bfp8_quantize4(v);
      *(v4f*)(y + (size_t)t * TILE_FLOATS + tid * 4u) = r;
    }
  }
#endif
}

extern "C" void kernel_launch(void* const* d_in, const int* in_sizes, int n_in,
                              void* d_out, int out_size, void* d_ws, size_t ws_size,
                              hipStream_t stream) {
  const float* x = (const float*)d_in[0];
  float* y = (float*)d_out;

  const unsigned n        = (unsigned)in_sizes[0];  // 8192*8192 = 64Mi
  const unsigned n_blocks = n / 128u;               // last dim divisible by 128
  const unsigned n_tiles  = (n_blocks + 7u) >> 3;
  unsigned grid = n_tiles < MAX_GRID ? n_tiles : MAX_GRID;
  if (grid == 0u) grid = 1u;

  bfp8_cast_kernel<<<dim3(grid), dim3(THREADS), 0, stream>>>(x, y, n_blocks);
}